// scale_and_CDF_33715493273920
// MI455X (gfx1250) — compile-verified
//
#include <hip/hip_runtime.h>
#include <math.h>

typedef __attribute__((ext_vector_type(2))) float v2f;
typedef __attribute__((ext_vector_type(8))) float v8f;

#define NBINS 32
#define DIM   64

// ---- workspace layout (float offsets) ----
// pv  : [32][64] float4 {v1=pdf[k], dvh=(pdf[k+1]-pdf[k])/es[k], F[k], 0}
// mesh: [32] float
// be  : [64] float2 {b[d], exp(clip(logs[d]))}
// scalars: sum_ls, C1=(R-1)/X1L, ILR=1/ln(R)
#define WS_PV  0
#define WS_ME  (NBINS * DIM * 4)        // 8192
#define WS_BE  (WS_ME + NBINS)          // 8224
#define WS_SUM (WS_BE + 2 * DIM)        // 8352
#define WS_C1  (WS_SUM + 1)
#define WS_ILR (WS_SUM + 2)

__global__ void setup_kernel(const float* __restrict__ p, const float* __restrict__ b,
                             const float* __restrict__ logs, float* __restrict__ ws) {
  __shared__ float red[DIM];
  const int d = threadIdx.x;  // 0..63, one column per thread

  // mesh in double precision, matching numpy, then cast to f32
  double rpow[17];
  double rp = 1.0;
  #pragma unroll
  for (int i = 0; i <= 16; ++i) { rpow[i] = rp; rp *= 1.2; }
  const double x1L = 50.0 * (1.2 - 1.0) / (rpow[16] - 1.0);
  float meshf[33];
  #pragma unroll
  for (int i = 0; i <= 32; ++i) {
    int idx = i - 16;
    int a = idx >= 0 ? idx : -idx;
    double xr = (1.0 - rpow[a]) / (1.0 - 1.2);
    xr = (idx >= 0) ? x1L * xr : -x1L * xr;
    meshf[i] = (float)((xr + 50.0) / 100.0);
  }
  meshf[0] = 0.0f; meshf[32] = 1.0f;
  float es[32];
  #pragma unroll
  for (int i = 0; i < 32; ++i) es[i] = meshf[i + 1] - meshf[i];

  // per-column pdf normalization (fp32, like the reference)
  const float ls    = fminf(fmaxf(logs[d], -5.0f), 5.0f);
  const float expls = expf(ls);
  float ep[31];
  float S = 0.0f;
  #pragma unroll
  for (int i = 0; i < 31; ++i) {
    ep[i] = expf(p[i * DIM + d]);
    float w = (es[i] + es[i + 1]) * 0.5f;
    S += ep[i] * w;
  }
  const float scale = (1.0f - es[0]) / S;
  float pdf[33];
  pdf[0] = 1.0f; pdf[32] = 1.0f;
  #pragma unroll
  for (int i = 0; i < 31; ++i) pdf[i + 1] = ep[i] * scale;

  float4* pv = (float4*)(ws + WS_PV);
  float F = 0.0f;
  #pragma unroll
  for (int k = 0; k < 32; ++k) {
    float dvh = (pdf[k + 1] - pdf[k]) / es[k];   // exact same fp32 operands as per-element div
    pv[k * DIM + d] = make_float4(pdf[k], dvh, F, 0.0f);
    F += (pdf[k] + pdf[k + 1]) * 0.5f * es[k];
  }
  if (d < 32) ws[WS_ME + d] = meshf[d];
  ((float2*)(ws + WS_BE))[d] = make_float2(b[d], expls);

  red[d] = ls;
  __syncthreads();
  if (d == 0) {
    float s = 0.0f;
    for (int i = 0; i < DIM; ++i) s += red[i];
    ws[WS_SUM] = s;
    ws[WS_C1]  = (float)((1.2 - 1.0) / (x1L / 100.0));
    ws[WS_ILR] = (float)(1.0 / log(1.2));
  }
}

__global__ void __launch_bounds__(256) cdf_main_kernel(
    const float* __restrict__ x, const float* __restrict__ logdet_in,
    const float* __restrict__ ws, float* __restrict__ out,
    float* __restrict__ logdet_out, int nrows) {
  // padded to stride 65 so data-dependent bin gathers spread across bank groups
  __shared__ float4 s_pv[NBINS * 65];
  __shared__ float  s_mesh[NBINS];
  __shared__ float2 s_be[DIM];
  __shared__ float  s_scalar[3];

  const int tid = threadIdx.x;
  const float4* wpv = (const float4*)(ws + WS_PV);
  for (int i = tid; i < NBINS * DIM; i += 256) {
    int k = i >> 6, c = i & 63;
    s_pv[k * 65 + c] = wpv[i];
  }
  if (tid < NBINS) s_mesh[tid] = ws[WS_ME + tid];
  if (tid < DIM)   s_be[tid]   = ((const float2*)(ws + WS_BE))[tid];
  if (tid < 3)     s_scalar[tid] = ws[WS_SUM + tid];
  __syncthreads();

  const int lane = tid & 31;
  const int wv   = tid >> 5;
  const int row0 = (blockIdx.x * 8 + wv) * 16;
  if (row0 >= nrows) return;                 // wave-uniform: survivors keep EXEC all-ones

  // WMMA A-matrix (16x4 f32) lane mapping: lanes 0-15 -> row=lane, K=0,1 (cols 0-31)
  //                                        lanes 16-31 -> row=lane-16, K=2,3 (cols 32-63)
  const int m     = lane & 15;
  const int row   = row0 + m;
  const int cbase = (lane < 16) ? 0 : 32;
  const float C1  = s_scalar[1];
  const float ILR = s_scalar[2];

  const float4* x4 = (const float4*)x + (size_t)row * 16;
  float4*       o4 = (float4*)out + (size_t)row * 16;

  float a01[2];
  #pragma unroll
  for (int g = 0; g < 2; ++g) {
    float acc = 0.0f;
    #pragma unroll
    for (int q = 0; q < 4; ++q) {
      const int c0 = cbase + g * 16 + q * 4;
      float4 xv = x4[c0 >> 2];
      float xe[4] = {xv.x, xv.y, xv.z, xv.w};
      float ye[4];
      #pragma unroll
      for (int e = 0; e < 4; ++e) {
        const int c = c0 + e;
        float2 be = s_be[c];                       // same addr across half-wave: LDS broadcast
        float z  = (xe[e] + be.x) * be.y;          // actnorm
        float xs = (z + 50.0f) * 0.01f;            // map to [0,1]
        float t  = xs - 0.5f;
        float at = fabsf(t);
        float mfl = floorf(__logf(at * C1 + 1.0f) * ILR);
        int   mi  = (int)mfl;                      // mfl >= 0 always
        int   k   = (t >= 0.0f) ? (16 + mi) : (15 - mi);
        bool  cover = (k * (k - 31)) <= 0;
        int   kc  = k < 0 ? 0 : (k > 31 ? 31 : k);
        float4 pv = s_pv[kc * 65 + c];             // {v1, dvh, F_pre, -}
        float xm  = xs - s_mesh[kc];
        float yq  = pv.z + xm * xm * 0.5f * pv.y + xm * pv.x;
        float dl  = xm * pv.y + pv.x;
        ye[e] = (cover ? yq : xs) * 100.0f - 50.0f;
        acc  += cover ? __logf(dl) : 0.0f;         // log(1)==0 branch folded into select
      }
      o4[c0 >> 2] = make_float4(ye[0], ye[1], ye[2], ye[3]);
    }
    a01[g] = acc;
  }

  // Row reduction on the matrix unit: C[m][n] = sum_k A[m][k] with B = ones(4x16).
  v2f A; A.x = a01[0]; A.y = a01[1];
  v2f B; B.x = 1.0f;   B.y = 1.0f;
  v8f Cz = {0.f, 0.f, 0.f, 0.f, 0.f, 0.f, 0.f, 0.f};
  v8f Cr = __builtin_amdgcn_wmma_f32_16x16x4_f32(
      /*neg_a=*/false, A, /*neg_b=*/false, B,
      /*c_mod=*/(short)0, Cz, /*reuse_a=*/false, /*reuse_b=*/false);

  // C layout: VGPR j, lanes 0-15 -> M=j; lanes 16-31 -> M=j+8 (all N identical here).
  // Lane 0 holds rows row0..row0+7 in Cr[0..7]; lane 16 holds rows row0+8..row0+15.
  // Single predicated path, vectorized 2x b128 load + 2x b128 store per active lane.
  const float sls = s_scalar[0];
  if ((lane & 15) == 0) {
    const int base = row0 + ((lane >> 4) << 3);    // lane 0 -> row0, lane 16 -> row0+8
    const float4* li = (const float4*)(logdet_in + base);
    float4*       lo = (float4*)(logdet_out + base);
    float4 l0 = li[0];
    float4 l1 = li[1];
    lo[0] = make_float4(l0.x + sls + Cr[0], l0.y + sls + Cr[1],
                        l0.z + sls + Cr[2], l0.w + sls + Cr[3]);
    lo[1] = make_float4(l1.x + sls + Cr[4], l1.y + sls + Cr[5],
                        l1.z + sls + Cr[6], l1.w + sls + Cr[7]);
  }
}

extern "C" void kernel_launch(void* const* d_in, const int* in_sizes, int n_in,
                              void* d_out, int out_size, void* d_ws, size_t ws_size,
                              hipStream_t stream) {
  const float* x    = (const float*)d_in[0];
  const float* ld   = (const float*)d_in[1];
  const float* p    = (const float*)d_in[2];
  const float* b    = (const float*)d_in[3];
  const float* logs = (const float*)d_in[4];

  float* out = (float*)d_out;
  float* ws  = (float*)d_ws;

  const int nd    = in_sizes[0];        // N*64
  const int nrows = nd / DIM;           // N
  float* logdet_out = out + nd;         // outputs concatenated: (N*64) then (N)

  setup_kernel<<<1, DIM, 0, stream>>>(p, b, logs, ws);
  const int nblocks = (nrows + 127) / 128;   // 8 waves/block * 16 rows/wave
  cdf_main_kernel<<<nblocks, 256, 0, stream>>>(x, ld, ws, out, logdet_out, nrows);
}